// FusedExpertsNetwork_75402445849115
// MI455X (gfx1250) — compile-verified
//
#include <hip/hip_runtime.h>

// MoE expert FFN: y = relu(x @ W1^T + b1) @ W2 + b2, batched over 8 experts.
// f32-exact path: V_WMMA_F32_16X16X4_F32 + double-buffered async global->LDS
// staging (GLOBAL_LOAD_ASYNC_TO_LDS_B128 / ASYNCcnt), one barrier per panel.

typedef float v2f __attribute__((ext_vector_type(2)));
typedef float v8f __attribute__((ext_vector_type(8)));

constexpr int kE   = 8;
constexpr int kCAP = 2048;
constexpr int kD   = 1024;   // model dim
constexpr int kH   = 4096;   // hidden dim

constexpr int TM  = 32;      // tokens per workgroup
constexpr int HK  = 64;      // hidden units per chunk
constexpr int KB  = 64;      // K-panel (phase A, over D)
constexpr int KB2 = 4;       // K-panel (phase B, over H-chunk)

constexpr int S68 = 68;      // padded row stride (even->b64 ok, bank stride 4)
constexpr int W2S = 1032;    // W2 panel row stride (row-pair bank offset 16)

constexpr int XS_SZ  = TM * S68;        // 2176 f32
constexpr int W1_SZ  = HK * S68;        // 4352 f32
constexpr int ABUF   = XS_SZ + W1_SZ;   // 6528 f32 per phase-A buffer
constexpr int BSH_SZ = KB2 * W2S;       // 4128 f32 per phase-B buffer
constexpr int HS_OFF = 2 * ABUF;        // 13056 (A pair; B pair aliases it)
constexpr int LDS_F32 = HS_OFF + TM * S68;  // 15232 f32 = 60,928 B

__device__ __forceinline__ v8f wmma_f32(v2f a, v2f b, v8f c) {
  return __builtin_amdgcn_wmma_f32_16x16x4_f32(false, a, false, b, (short)0, c,
                                               false, false);
}

// Async global -> LDS, 16B per lane, tracked by ASYNCcnt.
__device__ __forceinline__ void asy128(const float* lds_p, const float* g_p) {
  unsigned loff = (unsigned)(uintptr_t)lds_p;  // LDS aperture: addr[31:0] = offset
  unsigned long long ga = (unsigned long long)(uintptr_t)g_p;
  asm volatile("global_load_async_to_lds_b128 %0, %1, off"
               :: "v"(loff), "v"(ga) : "memory");
}
__device__ __forceinline__ void wait_async0() {
  asm volatile("s_wait_asynccnt 0" ::: "memory");
}

extern "C" __global__ void __launch_bounds__(256)
moe_ffn_f32_wmma(const float* __restrict__ x,
                 const float* __restrict__ w1,
                 const float* __restrict__ b1,
                 const float* __restrict__ w2,
                 const float* __restrict__ b2,
                 float* __restrict__ out) {
  __shared__ float lds[LDS_F32];
  float* hs = lds + HS_OFF;   // [TM][S68] relu'd hidden chunk

  const int tid  = threadIdx.x;
  const int lane = tid & 31;
  const int wave = tid >> 5;        // 0..7
  const int l16  = lane & 15;
  const int hi   = lane >> 4;       // fragment half
  const int koff = hi << 1;         // K offset within fragment (0 or 2)

  const int bid = blockIdx.x;       // E * CAP/TM = 512 blocks
  const int e   = bid >> 6;         // 64 blocks per expert
  const int t0  = (bid & 63) * TM;

  const float* xe  = x  + ((size_t)e * kCAP + t0) * kD;
  const float* w1e = w1 + (size_t)e * kH * kD;
  const float* w2e = w2 + (size_t)e * kH * kD;
  const float* b1e = b1 + e * kH;
  const float* b2e = b2 + e * kD;
  float*       oe  = out + ((size_t)e * kCAP + t0) * kD;

  const int am0 = (wave & 1) * 16;   // phase-A token-row base
  const int an0 = (wave >> 1) * 16;  // phase-A hidden-col base in chunk

  // ---- async panel issue helpers (16B per lane, coalesced rows) ----
  auto issueA = [&](int buf, int kk, int h0) {
    float* xs  = lds + buf * ABUF;
    float* w1p = xs + XS_SZ;
#pragma unroll
    for (int it = 0; it < (TM * KB / 4) / 256; ++it) {     // 2
      int i = tid + it * 256;
      int m = i >> 4, k4 = (i & 15) << 2;
      asy128(&xs[m * S68 + k4], &xe[(size_t)m * kD + kk + k4]);
    }
#pragma unroll
    for (int it = 0; it < (HK * KB / 4) / 256; ++it) {     // 4
      int i = tid + it * 256;
      int n = i >> 4, k4 = (i & 15) << 2;
      // row-major: w1p[n][k] = w1e[h0+n][kk+k]  (B frag -> single b64)
      asy128(&w1p[n * S68 + k4], &w1e[(size_t)(h0 + n) * kD + kk + k4]);
    }
  };
  auto issueB = [&](int buf, int h0, int kk2) {
    float* bs = lds + buf * BSH_SZ;
#pragma unroll
    for (int it = 0; it < (KB2 * kD / 4) / 256; ++it) {    // 4
      int i = tid + it * 256;
      int r = i >> 8, c4 = (i & 255) << 2;
      asy128(&bs[r * W2S + c4], &w2e[(size_t)(h0 + kk2 + r) * kD + c4]);
    }
  };

  const v8f vzero = {0.f, 0.f, 0.f, 0.f, 0.f, 0.f, 0.f, 0.f};
  v8f cacc[2][8];
#pragma unroll
  for (int mi = 0; mi < 2; ++mi)
#pragma unroll
    for (int ni = 0; ni < 8; ++ni) cacc[mi][ni] = vzero;

  issueA(0, 0, 0);   // software-pipeline prologue

  for (int c = 0; c < kH / HK; ++c) {     // 64 hidden chunks
    const int h0 = c * HK;

    // ---------- Phase A: hacc = X * W1^T over K = D ----------
    v8f hacc = vzero;
    for (int ip = 0; ip < kD / KB; ++ip) {        // 16 panels
      wait_async0();        // own loads for panel ip landed
      __syncthreads();      // everyone's landed; prev buffer's readers done
      if (ip < kD / KB - 1) issueA((ip + 1) & 1, (ip + 1) * KB, h0);

      const float* xs  = lds + (ip & 1) * ABUF;
      const float* w1p = xs + XS_SZ;
#pragma unroll
      for (int k = 0; k < KB; k += 4) {
        v2f a = *(const v2f*)&xs[(am0 + l16) * S68 + k + koff];
        v2f b = *(const v2f*)&w1p[(an0 + l16) * S68 + k + koff];
        hacc = wmma_f32(a, b, hacc);
      }
    }

    // bias + relu, park chunk in LDS (C-layout: VGPR r -> row r + 8*hi)
    {
      const float bias1 = b1e[h0 + an0 + l16];
#pragma unroll
      for (int r = 0; r < 8; ++r) {
        float v = hacc[r] + bias1;
        v = v > 0.f ? v : 0.f;
        hs[(am0 + r + (hi << 3)) * S68 + an0 + l16] = v;
      }
    }

    __syncthreads();          // hs visible; phase-A buffers now reusable
    issueB(0, h0, 0);

    // ---------- Phase B: C += h_chunk * W2_chunk ----------
    for (int ip = 0; ip < HK / KB2; ++ip) {       // 16 panels
      wait_async0();
      __syncthreads();
      if (ip < HK / KB2 - 1) issueB((ip + 1) & 1, h0, (ip + 1) * KB2);

      const float* bs = lds + (ip & 1) * BSH_SZ;
      const int kk2 = ip * KB2;
      v2f a0 = *(const v2f*)&hs[(l16) * S68 + kk2 + koff];
      v2f a1 = *(const v2f*)&hs[(16 + l16) * S68 + kk2 + koff];
#pragma unroll
      for (int ni = 0; ni < 8; ++ni) {
        const int n = wave * 128 + ni * 16 + l16;
        v2f b;
        b.x = bs[(koff + 0) * W2S + n];
        b.y = bs[(koff + 1) * W2S + n];
        cacc[0][ni] = wmma_f32(a0, b, cacc[0][ni]);
        cacc[1][ni] = wmma_f32(a1, b, cacc[1][ni]);
      }
    }

    __syncthreads();          // all W2-panel reads done
    if (c + 1 < kH / HK) issueA(0, 0, h0 + HK);   // next chunk, panel 0
  }

  // ---------- Epilogue: + b2, store ----------
#pragma unroll
  for (int mi = 0; mi < 2; ++mi)
#pragma unroll
    for (int ni = 0; ni < 8; ++ni) {
      const int n = wave * 128 + ni * 16 + l16;
      const float bias2 = b2e[n];
#pragma unroll
      for (int r = 0; r < 8; ++r) {
        const int row = mi * 16 + r + (hi << 3);
        oe[(size_t)row * kD + n] = cacc[mi][ni][r] + bias2;
      }
    }
}

extern "C" void kernel_launch(void* const* d_in, const int* in_sizes, int n_in,
                              void* d_out, int out_size, void* d_ws, size_t ws_size,
                              hipStream_t stream) {
  (void)in_sizes; (void)n_in; (void)out_size; (void)d_ws; (void)ws_size;
  const float* x  = (const float*)d_in[0];
  const float* w1 = (const float*)d_in[1];
  const float* b1 = (const float*)d_in[2];
  const float* w2 = (const float*)d_in[3];
  const float* b2 = (const float*)d_in[4];
  float* out = (float*)d_out;

  dim3 grid(kE * (kCAP / TM));   // 512 workgroups
  dim3 block(256);               // 8 waves (wave32)
  hipLaunchKernelGGL(moe_ffn_f32_wmma, grid, block, 0, stream,
                     x, w1, b1, w2, b2, out);
}